// SingleRoIExtractor_9174050144296
// MI455X (gfx1250) — compile-verified
//
#include <hip/hip_runtime.h>
#include <math.h>

// SingleRoIExtractor (FPN RoIAlign, aligned=True) for MI455X / gfx1250.
//
// Roofline: ~0.5 GFLOP vs a few hundred MB of L2-resident gathers -> pure
// gather-latency/bandwidth problem; fp32 mandated; WMMA inapplicable.
// Design: one workgroup per RoI, lane = channel (all sample coordinates are
// wave-uniform -> SALU/scalar-float pipeline), bilinear weights precomputed
// once per block into LDS with validity folded in, full-patch
// global_prefetch_b8 warming, non-temporal stores for the write-once output.

#define OUT_SIZE 7
#define SR 2
#define P (OUT_SIZE * SR)   // 14 samples per axis
#define C_CH 256
#define FINEST 56.0f

__global__ __launch_bounds__(256) void roi_align_fpn_kernel(
    const float* __restrict__ f0, const float* __restrict__ f1,
    const float* __restrict__ f2, const float* __restrict__ f3,
    const float* __restrict__ rois, float* __restrict__ out, int N)
{
    const int n   = blockIdx.x;
    if (n >= N) return;
    const int tid = threadIdx.x;

    // Per-axis sample tables (wave-uniform, shared across all 256 channels)
    __shared__ int   sx0[P], sx1[P];      // clamped column indices
    __shared__ int   sy0[P], sy1[P];      // row offsets, pre-multiplied by W
    __shared__ float shx[P], slx[P];      // x weights (validity folded in)
    __shared__ float shy[P], sly[P];      // y weights (validity folded in)

    // ---- RoI params (uniform across the block -> scalar pipeline) ----
    const float rb  = rois[n * 5 + 0];
    const float rx1 = rois[n * 5 + 1];
    const float ry1 = rois[n * 5 + 2];
    const float rx2 = rois[n * 5 + 3];
    const float ry2 = rois[n * 5 + 4];
    const int   b   = (int)rb;
    const float rw  = rx2 - rx1;
    const float rh  = ry2 - ry1;

    // Level assignment: clip(floor(log2(sqrt(w*h)/56 + 1e-6)), 0, 3)
    const float sc  = sqrtf(rw * rh);
    int lvl = (int)floorf(log2f(sc * (1.0f / FINEST) + 1e-6f));
    lvl = min(max(lvl, 0), 3);

    const float* feat;
    int H;                                   // H == W for every level
    switch (lvl) {
        case 0:  feat = f0; H = 256; break;  // stride 4
        case 1:  feat = f1; H = 128; break;  // stride 8
        case 2:  feat = f2; H = 64;  break;  // stride 16
        default: feat = f3; H = 32;  break;  // stride 32
    }
    const float fscale = 1.0f / (float)(4 << lvl);
    const float Hf     = (float)H;

    // ---- Precompute the 14 x- and 14 y-sample entries (one thread each) ----
    if (tid < P) {
        const int   i     = tid;
        const float start = rx1 * fscale - 0.5f;
        const float binw  = rw * fscale * (1.0f / OUT_SIZE);
        const float xs    = start + binw * (((float)i + 0.5f) * (1.0f / SR));
        const float vm    = ((xs > -1.0f) && (xs < Hf)) ? 1.0f : 0.0f;
        const float xc    = fminf(fmaxf(xs, 0.0f), Hf - 1.0f);
        const float x0f   = floorf(xc);
        const float lx    = xc - x0f;
        const int   x0    = (int)x0f;
        sx0[i] = x0;
        sx1[i] = min(x0 + 1, H - 1);
        slx[i] = lx * vm;
        shx[i] = (1.0f - lx) * vm;
    } else if (tid >= 32 && tid < 32 + P) {
        const int   i     = tid - 32;
        const float start = ry1 * fscale - 0.5f;
        const float binh  = rh * fscale * (1.0f / OUT_SIZE);
        const float ys    = start + binh * (((float)i + 0.5f) * (1.0f / SR));
        const float vm    = ((ys > -1.0f) && (ys < Hf)) ? 1.0f : 0.0f;
        const float yc    = fminf(fmaxf(ys, 0.0f), Hf - 1.0f);
        const float y0f   = floorf(yc);
        const float ly    = yc - y0f;
        const int   y0    = (int)y0f;
        sy0[i] = y0 * H;                       // pre-scaled row offset
        sy1[i] = min(y0 + 1, H - 1) * H;
        sly[i] = ly * vm;
        shy[i] = (1.0f - ly) * vm;
    }
    __syncthreads();

    // ---- Per-channel gather + bilinear + 2x2 mean ----
    const int    c    = tid;                   // lane = channel
    const float* base = feat + (size_t)(b * C_CH + c) * (size_t)(H * H);
    float*       o    = out + ((size_t)n * C_CH + c) * (OUT_SIZE * OUT_SIZE);

    // Warm WGP$ for the whole sample patch (gfx1250 global_prefetch_b8):
    // one touch per 64B line (16 floats) across [xlo, xhi] for both bilinear
    // row sets. Wide level-0 RoIs can span ~120 columns, so cover adaptively.
    const int xlo = sx0[0], xhi = sx1[P - 1];
    for (int i = 0; i < P; ++i) {
        const float* r0 = base + sy0[i];
        const float* r1 = base + sy1[i];
        for (int xp = xlo; xp <= xhi; xp += 16) {
            __builtin_prefetch(r0 + xp, 0, 3);
            __builtin_prefetch(r1 + xp, 0, 3);
        }
        __builtin_prefetch(r0 + xhi, 0, 3);
        __builtin_prefetch(r1 + xhi, 0, 3);
    }

    for (int ph = 0; ph < OUT_SIZE; ++ph) {
        const int ia = 2 * ph, ib = 2 * ph + 1;
        const float* rA0 = base + sy0[ia];
        const float* rA1 = base + sy1[ia];
        const float* rB0 = base + sy0[ib];
        const float* rB1 = base + sy1[ib];
        const float hyA = shy[ia], lyA = sly[ia];
        const float hyB = shy[ib], lyB = sly[ib];

        #pragma unroll
        for (int pw = 0; pw < OUT_SIZE; ++pw) {
            float acc = 0.0f;
            #pragma unroll
            for (int k = 0; k < SR; ++k) {
                const int   ix = 2 * pw + k;
                const int   x0 = sx0[ix], x1 = sx1[ix];
                const float hx = shx[ix], lx = slx[ix];
                const float vA = hyA * (hx * rA0[x0] + lx * rA0[x1])
                               + lyA * (hx * rA1[x0] + lx * rA1[x1]);
                const float vB = hyB * (hx * rB0[x0] + lx * rB0[x1])
                               + lyB * (hx * rB1[x0] + lx * rB1[x1]);
                acc += vA + vB;
            }
            // Output (51 MB) is write-once: non-temporal store keeps L2 for
            // the feature pyramids every block re-gathers from.
            __builtin_nontemporal_store(acc * 0.25f, &o[ph * OUT_SIZE + pw]);
        }
    }
}

extern "C" void kernel_launch(void* const* d_in, const int* in_sizes, int n_in,
                              void* d_out, int out_size, void* d_ws, size_t ws_size,
                              hipStream_t stream)
{
    const float* f0   = (const float*)d_in[0];
    const float* f1   = (const float*)d_in[1];
    const float* f2   = (const float*)d_in[2];
    const float* f3   = (const float*)d_in[3];
    const float* rois = (const float*)d_in[4];
    float*       out  = (float*)d_out;

    const int N = in_sizes[4] / 5;   // rois are [N, 5]

    roi_align_fpn_kernel<<<N, 256, 0, stream>>>(f0, f1, f2, f3, rois, out, N);
}